// InfoNCELoss_31542239822269
// MI455X (gfx1250) — compile-verified
//
#include <hip/hip_runtime.h>
#include <math.h>

// ---------------------------------------------------------------------------
// InfoNCE (NT-Xent) loss, B=4096 D=128 T=0.5, fp32.
//
// loss = (1/8192) * sum_r [ LSE_{c != r}( (z_r . z_c) * 2 ) - pos_sim[r mod B] ]
// where z = concat(z_i, z_j)  (8192 x 128).
//
// MI455X reasoning: 17.2 GFLOP fp32 GEMM over 4 MB -> L2-resident, compute
// bound on the matrix pipes; fp32 exactness => V_WMMA_F32_16X16X4_F32.
//  * B operands for a whole 16x16 tile loaded into a register array first
//    (batched ds loads, single s_wait_dscnt), then 32 WMMAs back-to-back.
//  * Next B tile staged with GLOBAL_LOAD_ASYNC_TO_LDS_B128 (ASYNCcnt) so the
//    DMA overlaps the WMMA burst; one s_wait_asynccnt + one barrier per tile.
//    Builtin signature (from hipcc diagnostic): (int4 AS1*, int4 AS3*, Ii, Ii).
// ---------------------------------------------------------------------------

#define BROWS 4096
#define DIM   128
#define NTOT  8192              // 2*B
#define INV_T 2.0f
#define NEG_BIG (-1.0e30f)
#define CS    8                 // column splits (blockIdx.x)
#define COLS_PER_BLK (NTOT / CS)        // 1024
#define ROWS_PER_BLK 128
#define TPB   256               // 8 wave32
#define BPAD  132               // padded LDS row stride (floats)
#define KSTEPS (DIM / 4)        // 32 WMMA (K=4) per 16x16 tile

typedef float v2f __attribute__((ext_vector_type(2)));
typedef float v8f __attribute__((ext_vector_type(8)));

#if defined(__gfx1250__) && __has_builtin(__builtin_amdgcn_global_load_async_to_lds_b128)
#define USE_ASYNC 1
#else
#define USE_ASYNC 0
#endif

#if USE_ASYNC
#if __has_builtin(__builtin_amdgcn_s_wait_asynccnt)
#define WAIT_ASYNC() __builtin_amdgcn_s_wait_asynccnt(0)
#else
#define WAIT_ASYNC() asm volatile("s_wait_asynccnt 0x0" ::: "memory")
#endif
// Builtin parameter types per hipcc diagnostic:
//   arg0: int __vector_size__(16) __device__ *   (global, AS1)
//   arg1: int __vector_size__(16) __shared__ *   (LDS, AS3)
typedef int v4i_gcc __attribute__((vector_size(16)));
typedef __attribute__((address_space(1))) v4i_gcc* gv4i_p;
typedef __attribute__((address_space(3))) v4i_gcc* lv4i_p;
#endif

__device__ __forceinline__ const float* zrow(const float* zi, const float* zj, int r) {
    return (r < BROWS) ? (zi + (size_t)r * DIM) : (zj + (size_t)(r - BROWS) * DIM);
}

#if USE_ASYNC
// Per-thread: async-copy 8 consecutive floats (two B128 transfers) from a
// global row directly into LDS. IOFFSET is added to both addresses (ISA 8.4.4).
__device__ __forceinline__ void stage_async(const float* src, float* lds_dst) {
    gv4i_p g = (gv4i_p)src;
    lv4i_p l = (lv4i_p)lds_dst;
    __builtin_amdgcn_global_load_async_to_lds_b128(g, l, 0, 0);
    __builtin_amdgcn_global_load_async_to_lds_b128(g, l, 16, 0);
}
#endif

// --------------------------- pos_sim[r] = (z_i[r].z_j[r]) * 2 ---------------
__global__ void __launch_bounds__(TPB)
pos_kernel(const float* __restrict__ zi, const float* __restrict__ zj,
           float* __restrict__ pos) {
    const int lane = threadIdx.x & 31;
    const int wave = threadIdx.x >> 5;
    const int r = blockIdx.x * (TPB / 32) + wave;
    const float4 a = *(const float4*)(zi + (size_t)r * DIM + lane * 4);
    const float4 b = *(const float4*)(zj + (size_t)r * DIM + lane * 4);
    float d = a.x * b.x + a.y * b.y + a.z * b.z + a.w * b.w;
#pragma unroll
    for (int off = 16; off > 0; off >>= 1) d += __shfl_xor(d, off);
    if (lane == 0) pos[r] = d * INV_T;
}

// ------------------ main: WMMA GEMM + streaming logsumexp -------------------
// block (split, by): rows [by*128, by*128+128), cols [split*1024, +1024)
// wave w: 16-row stripe; emits partial (max, sumexp) per row.
__global__ void __launch_bounds__(TPB)
simlse_kernel(const float* __restrict__ zi, const float* __restrict__ zj,
              float* __restrict__ ws_m, float* __restrict__ ws_s) {
    __shared__ float btile[2][16 * BPAD];

    const int tid   = threadIdx.x;
    const int lane  = tid & 31;
    const int wave  = tid >> 5;
    const int lo    = lane & 15;   // column slot / A row within stripe
    const int hi    = lane >> 4;   // selects K pair {0,1} vs {2,3}
    const int split = blockIdx.x;
    const int rBase = blockIdx.y * ROWS_PER_BLK + wave * 16;
    const int jBeg  = split * COLS_PER_BLK;

    // A stripe in registers, WMMA 16x4 fp32 layout, pre-scaled by 1/T.
    // lane: row = rBase+lo, a[t] = {A[row][4t+2hi], A[row][4t+2hi+1]} * 2
    const float* arow = zrow(zi, zj, rBase + lo);
    v2f a[KSTEPS];
#pragma unroll
    for (int t = 0; t < KSTEPS; ++t) {
        v2f av = *(const v2f*)(arow + 4 * t + 2 * hi);
        a[t] = av * INV_T;
    }

    // Lane-local streaming LSE state: 8 rows per lane (row = rBase + v + 8*hi),
    // each lane covers columns congruent to lo (mod 16) of this block's range.
    float m[8], s[8];
#pragma unroll
    for (int v = 0; v < 8; ++v) { m[v] = NEG_BIG; s[v] = 0.0f; }

    // Cooperative B-tile staging: 16 rows x 128 floats = one contiguous 8KB
    // region of z; thread -> 8 consecutive floats.
    const int trow = tid >> 4;         // 0..15
    const int tcol = (tid & 15) * 8;   // 0..120
    float* myDst0 = &btile[0][trow * BPAD + tcol];
    float* myDst1 = &btile[1][trow * BPAD + tcol];

    // stage tile 0 into buffer 0
#if USE_ASYNC
    stage_async(zrow(zi, zj, jBeg + trow) + tcol, myDst0);
    WAIT_ASYNC();
#else
    {
        const float* src = zrow(zi, zj, jBeg + trow) + tcol;
        *(float4*)(myDst0)     = *(const float4*)(src);
        *(float4*)(myDst0 + 4) = *(const float4*)(src + 4);
    }
#endif
    __syncthreads();

    const int NT = COLS_PER_BLK / 16;   // 64 column tiles
    for (int jt = 0; jt < NT; ++jt) {
        const int cb = jt & 1;
        const int j0 = jBeg + jt * 16;
        const bool pref = (jt + 1 < NT);

#if USE_ASYNC
        // Kick off the DMA for the next tile; it overlaps the WMMA burst.
        // Buffer cb^1 was last read in iteration jt-1 and all waves passed
        // that iteration's barrier before reaching this point.
        if (pref)
            stage_async(zrow(zi, zj, j0 + 16 + trow) + tcol,
                        cb ? myDst0 : myDst1);
#else
        float4 p0, p1;
        if (pref) {
            const float* src = zrow(zi, zj, j0 + 16 + trow) + tcol;
            p0 = *(const float4*)(src);
            p1 = *(const float4*)(src + 4);
        }
#endif

        // Fetch the whole tile's B operands into registers first (batched DS
        // loads, one dscnt wait), then run the 32 WMMAs back-to-back.
        // B layout mirrors A: lanes0-15 K={4t,4t+1}, lanes16-31 K={4t+2,4t+3},
        // N = lo. Padded stride keeps all 32 lanes on distinct bank pairs.
        v2f b[KSTEPS];
#pragma unroll
        for (int t = 0; t < KSTEPS; ++t)
            b[t] = *(const v2f*)&btile[cb][lo * BPAD + 4 * t + 2 * hi];

        v8f acc = {};
#pragma unroll
        for (int t = 0; t < KSTEPS; ++t)
            acc = __builtin_amdgcn_wmma_f32_16x16x4_f32(
                false, a[t], false, b[t], (short)0, acc, false, false);

        // Streaming LSE update; exclude self column (global col == global row).
        // acc[v] is sim for (row rBase+v+8*hi, col j0+lo); excluded iff v==dd.
        const int dd = j0 + lo - rBase - 8 * hi;
#pragma unroll
        for (int v = 0; v < 8; ++v) {
            float x = acc[v];
            const bool ex = (v == dd);
            x = ex ? NEG_BIG : x;
            const float nm = fmaxf(m[v], x);
            const float e  = ex ? 0.0f : __expf(x - nm);
            s[v] = s[v] * __expf(m[v] - nm) + e;
            m[v] = nm;
        }

#if USE_ASYNC
        if (pref) {
            WAIT_ASYNC();      // next tile fully landed in LDS (this wave)
            __syncthreads();   // ... and in every wave; also fences compute(jt)
        }
#else
        if (pref) {
            __syncthreads();   // all waves done reading buffer cb^1
            float* dst = cb ? myDst0 : myDst1;
            *(float4*)(dst)     = p0;
            *(float4*)(dst + 4) = p1;
            __syncthreads();   // tile visible before next compute
        }
#endif
    }

    // Merge the 16 column slots within each 16-lane half (xor masks < 16
    // never cross the half boundary), then lane lo==0 of each half writes
    // the partial (m, s) for row rBase + v + 8*hi.
#pragma unroll
    for (int v = 0; v < 8; ++v) {
        float mv = m[v], sv = s[v];
#pragma unroll
        for (int off = 1; off < 16; off <<= 1) {
            const float om = __shfl_xor(mv, off);
            const float os = __shfl_xor(sv, off);
            const float nm = fmaxf(mv, om);
            sv = sv * __expf(mv - nm) + os * __expf(om - nm);
            mv = nm;
        }
        if (lo == 0) {
            const int row = rBase + v + 8 * hi;
            ws_m[(size_t)split * NTOT + row] = mv;
            ws_s[(size_t)split * NTOT + row] = sv;
        }
    }
}

// --------- finalize: merge splits, subtract pos, deterministic reduce -------
__global__ void __launch_bounds__(TPB)
finalize_kernel(const float* __restrict__ pos,
                const float* __restrict__ ws_m, const float* __restrict__ ws_s,
                float* __restrict__ out) {
    __shared__ float red[TPB];
    float local = 0.0f;
    for (int row = threadIdx.x; row < NTOT; row += TPB) {
        float m = NEG_BIG, s = 0.0f;
#pragma unroll
        for (int c = 0; c < CS; ++c) {       // fixed order -> deterministic
            const float om = ws_m[(size_t)c * NTOT + row];
            const float os = ws_s[(size_t)c * NTOT + row];
            const float nm = fmaxf(m, om);
            s = s * __expf(m - nm) + os * __expf(om - nm);
            m = nm;
        }
        local += (m + __logf(s)) - pos[row & (BROWS - 1)];
    }
    red[threadIdx.x] = local;
    __syncthreads();
    for (int off = TPB / 2; off > 0; off >>= 1) {
        if (threadIdx.x < off) red[threadIdx.x] += red[threadIdx.x + off];
        __syncthreads();
    }
    if (threadIdx.x == 0) out[0] = red[0] * (1.0f / (float)NTOT);
}

// ---------------------------------------------------------------------------
extern "C" void kernel_launch(void* const* d_in, const int* in_sizes, int n_in,
                              void* d_out, int out_size, void* d_ws, size_t ws_size,
                              hipStream_t stream) {
    (void)in_sizes; (void)n_in; (void)out_size; (void)ws_size;
    const float* zi = (const float*)d_in[0];
    const float* zj = (const float*)d_in[1];

    float* ws   = (float*)d_ws;
    float* pos  = ws;                         // 4096 floats
    float* ws_m = ws + BROWS;                 // CS * 8192 floats
    float* ws_s = ws_m + (size_t)CS * NTOT;   // CS * 8192 floats
    // total workspace: (4096 + 2*8*8192) * 4 = 528 KB

    pos_kernel<<<BROWS / (TPB / 32), TPB, 0, stream>>>(zi, zj, pos);

    dim3 grid(CS, NTOT / ROWS_PER_BLK);       // 8 x 64 = 512 blocks
    simlse_kernel<<<grid, TPB, 0, stream>>>(zi, zj, ws_m, ws_s);

    finalize_kernel<<<1, TPB, 0, stream>>>(pos, ws_m, ws_s, (float*)d_out);
}